// AFNO3DLoRA_14259291423326
// MI455X (gfx1250) — compile-verified
//
#include <hip/hip_runtime.h>
#include <hip/hip_bf16.h>
#include <math.h>

// ---------------------------------------------------------------------------
// AFNO3D + LoRA for MI455X (gfx1250), truncated-DFT formulation.
// Every FFT stage is a small GEMM: TDM (tensor_load_to_lds) stages the B tile
// into LDS, V_WMMA_F32_16X16X4_F32 does the math, fp32 end to end.
// ---------------------------------------------------------------------------

typedef __attribute__((ext_vector_type(2))) float v2f;
typedef __attribute__((ext_vector_type(8))) float v8f;
typedef unsigned int u32x4 __attribute__((ext_vector_type(4)));
typedef int i32x4 __attribute__((ext_vector_type(4)));
typedef int i32x8 __attribute__((ext_vector_type(8)));

#define HAS_WMMA_F32X4 __has_builtin(__builtin_amdgcn_wmma_f32_16x16x4_f32)
#define HAS_TDM (__has_builtin(__builtin_amdgcn_tensor_load_to_lds) && \
                 __has_builtin(__builtin_amdgcn_s_wait_tensorcnt))

__device__ __forceinline__ v8f wmma_f32_k4(v2f a, v2f b, v8f c) {
#if HAS_WMMA_F32X4
  return __builtin_amdgcn_wmma_f32_16x16x4_f32(
      /*neg_a=*/false, a, /*neg_b=*/false, b,
      /*c_mod=*/(short)0, c, /*reuse_a=*/false, /*reuse_b=*/false);
#else
  c[0] += a.x * b.x + a.y * b.y;   // keep-alive fallback (histogram: wmma==0)
  return c;
#endif
}

// ---------------------------------------------------------------------------
// TDM: stage a 2D tile (rowElems f32 per line, nLines lines, line stride
// strideElems f32) from global into LDS. Single-wave issue; caller must
// s_wait_tensorcnt(0) in the issuing wave, then barrier.
// D# layout per CDNA5 ISA (group0 128b, group1 256b).
// This toolchain exposes the 6-arg builtin:
//   (u32x4 g0, i32x8 g1, i32x4 g2, i32x4 g3, i32x8 gx, i32 cpol)
// ---------------------------------------------------------------------------
#if HAS_TDM
__device__ __forceinline__ void tdm_load_2d(const float* gsrc, void* lds,
                                            unsigned rowElems, unsigned nLines,
                                            unsigned strideElems) {
  const unsigned long long ga = (unsigned long long)(uintptr_t)gsrc;
  const unsigned ldsOff = (unsigned)(uintptr_t)lds;  // low 32b of LDS aperture ptr
  u32x4 g0;
  g0.x = 1u;                                         // count=1, user descriptor
  g0.y = ldsOff;                                     // lds_addr (bytes)
  g0.z = (unsigned)(ga & 0xffffffffu);               // global_addr[31:0]
  g0.w = (unsigned)((ga >> 32) & 0x01ffffffu) | (2u << 30);  // ga[56:32] | type=2
  i32x8 g1;
  g1[0] = 0x00020000;                                // wg_mask=0, data_size=2 (4B)
  g1[1] = (int)((rowElems & 0xffffu) << 16);         // tensor_dim0[15:0]
  g1[2] = (int)((rowElems >> 16) & 0xffffu) |        // tensor_dim0[31:16]
          (int)((nLines & 0xffffu) << 16);           // tensor_dim1[15:0]
  g1[3] = (int)((rowElems & 0xffffu) << 16);         // tensor_dim1 hi=0, tile_dim0
  g1[4] = (int)(nLines & 0xffffu);                   // tile_dim1 (tile_dim2=0)
  g1[5] = (int)strideElems;                          // tensor_dim0_stride[31:0]
  g1[6] = 0;                                         // stride0 hi, stride1 lo
  g1[7] = 0;
  const i32x4 z4 = {0, 0, 0, 0};
  const i32x8 z8 = {0, 0, 0, 0, 0, 0, 0, 0};
  __builtin_amdgcn_tensor_load_to_lds(g0, g1, z4, z4, z8, 0);
}
#endif

// Shared WMMA inner loop: B tile staged in LDS as [k][64] f32.
__device__ __forceinline__ v8f gemm_acc(const float* __restrict__ Arow,
                                        const float* bt, int ncol, int klen,
                                        int half) {
  v8f acc = {};
  for (int k0 = 0; k0 < klen; k0 += 4) {
    const int k1 = k0 + 2 * half;
    v2f a, b;
    a.x = Arow[k1];
    a.y = Arow[k1 + 1];
    b.x = bt[k1 * 64 + ncol];
    b.y = bt[(k1 + 1) * 64 + ncol];
    acc = wmma_f32_k4(a, b, acc);
  }
  return acc;
}

// problem constants
constexpr int Bn = 4, Hn = 64, Wn = 64, Cn = 64;
constexpr int Mm = 32, Tt = 8;

// workspace layout (float offsets)
constexpr int OFF_FL  = 0;          // [16][64]   fwd L-DFT rows (2t+p)
constexpr int OFF_AF  = 1024;       // [64][128]  fwd W/H complex DFT realified
constexpr int OFF_AI  = 9216;       // [128][64]  inv W/H complex DFT realified
constexpr int OFF_GL  = 17408;      // [64][16]   inv L (hermitian irfft)
constexpr int OFF_W1R = 18432;      // [8][8][8]
constexpr int OFF_W1I = 18944;
constexpr int OFF_W2R = 19456;
constexpr int OFF_W2I = 19968;
constexpr int OFF_B1R = 20480;      // [64] each
constexpr int OFF_B1I = 20544;
constexpr int OFF_B2R = 20608;
constexpr int OFF_B2I = 20672;
constexpr size_t OFF_S1 = 32768;                 // [B][H][W][16][C]  67MB (also S5)
constexpr size_t OFF_S2 = OFF_S1 + 16777216;     // [B][H][T][64][C]  33.5MB (also S4)
constexpr size_t OFF_S3 = OFF_S2 + 8388608;      // [B][T][M][64][C]  16.8MB

// ---------------------------------------------------------------------------
// Tables + LoRA fold:  Weff = Wbase + (alpha/r) * A@B
// ---------------------------------------------------------------------------
__global__ __launch_bounds__(256) void afno_init_tables(
    const float* __restrict__ w1, const float* __restrict__ b1,
    const float* __restrict__ w2, const float* __restrict__ b2,
    const float* __restrict__ la1rA, const float* __restrict__ la1rB,
    const float* __restrict__ la1iA, const float* __restrict__ la1iB,
    const float* __restrict__ la2rA, const float* __restrict__ la2rB,
    const float* __restrict__ la2iA, const float* __restrict__ la2iB,
    float* __restrict__ ws) {
  const float W64 = 0.09817477042468103f;  // 2*pi/64
  for (int idx = threadIdx.x; idx < 1024; idx += 256) {   // FL [16][64]
    int t2 = idx >> 6, l = idx & 63, t = t2 >> 1;
    float ang = -W64 * (float)(t * l);
    ws[OFF_FL + idx] = ((t2 & 1) ? sinf(ang) : cosf(ang)) * 0.125f;
  }
  for (int idx = threadIdx.x; idx < 8192; idx += 256) {   // AF [64][128]
    int r = idx >> 7, k = idx & 127;
    int m = r >> 1, w = k >> 1;
    float ang = -W64 * (float)(m * w);
    float cr = cosf(ang) * 0.125f, ci = sinf(ang) * 0.125f;
    ws[OFF_AF + idx] = (!(r & 1)) ? ((k & 1) ? -ci : cr) : ((k & 1) ? cr : ci);
  }
  for (int idx = threadIdx.x; idx < 8192; idx += 256) {   // AI [128][64]
    int r = idx >> 6, k = idx & 63;
    int h = r >> 1, kk = k >> 1;
    float ang = W64 * (float)(h * kk);
    float cr = cosf(ang) * 0.125f, ci = sinf(ang) * 0.125f;
    ws[OFF_AI + idx] = (!(r & 1)) ? ((k & 1) ? -ci : cr) : ((k & 1) ? cr : ci);
  }
  for (int idx = threadIdx.x; idx < 1024; idx += 256) {   // GL [64][16]
    int l = idx >> 4, t2 = idx & 15, t = t2 >> 1;
    float alpha = ((t == 0) ? 1.0f : 2.0f) * 0.125f;
    float ang = W64 * (float)(t * l);
    ws[OFF_GL + idx] = (t2 & 1) ? -alpha * sinf(ang) : alpha * cosf(ang);
  }
  for (int idx = threadIdx.x; idx < 512; idx += 256) {    // LoRA fold
    int n = idx >> 6, i = (idx >> 3) & 7, o = idx & 7;
    float s1r = 0.f, s1i = 0.f, s2r = 0.f, s2i = 0.f;
    for (int r = 0; r < 8; ++r) {
      s1r += la1rA[(n * 8 + i) * 8 + r] * la1rB[(n * 8 + r) * 8 + o];
      s1i += la1iA[(n * 8 + i) * 8 + r] * la1iB[(n * 8 + r) * 8 + o];
      s2r += la2rA[(n * 8 + i) * 8 + r] * la2rB[(n * 8 + r) * 8 + o];
      s2i += la2iA[(n * 8 + i) * 8 + r] * la2iB[(n * 8 + r) * 8 + o];
    }
    ws[OFF_W1R + idx] = w1[idx]       + 0.125f * s1r;
    ws[OFF_W1I + idx] = w1[512 + idx] + 0.125f * s1i;
    ws[OFF_W2R + idx] = w2[idx]       + 0.125f * s2r;
    ws[OFF_W2I + idx] = w2[512 + idx] + 0.125f * s2i;
  }
  for (int idx = threadIdx.x; idx < 64; idx += 256) {
    ws[OFF_B1R + idx] = b1[idx];
    ws[OFF_B1I + idx] = b1[64 + idx];
    ws[OFF_B2R + idx] = b2[idx];
    ws[OFF_B2I + idx] = b2[64 + idx];
  }
}

// Macro: TDM-stage a tile into LDS (fallback: cooperative float4 copy of
// strided lines), then barrier.
#if HAS_TDM
#define STAGE_TILE_2D(DST, SRC, ROWE, NLINES, STRIDE, NTHR)          \
  do {                                                               \
    if (threadIdx.x < 32) {                                          \
      tdm_load_2d((SRC), (DST), (ROWE), (NLINES), (STRIDE));         \
      __builtin_amdgcn_s_wait_tensorcnt(0);                          \
    }                                                                \
    __syncthreads();                                                 \
  } while (0)
#else
#define STAGE_TILE_2D(DST, SRC, ROWE, NLINES, STRIDE, NTHR)          \
  do {                                                               \
    const int nf4 = (ROWE) / 4;                                      \
    for (int j = threadIdx.x; j < (NLINES)*nf4; j += (NTHR)) {       \
      int line = j / nf4, q = j % nf4;                               \
      ((float4*)(DST))[line * nf4 + q] =                             \
          ((const float4*)((SRC) + (size_t)line * (STRIDE)))[q];     \
    }                                                                \
    __syncthreads();                                                 \
  } while (0)
#endif

// ---------------------------------------------------------------------------
// Stage 1: truncated rDFT along L. blocks = B*H*W, 128 thr (4 waves).
// [16 x 64] @ [64 x 64] -> S1[bhw][t2][c]
// ---------------------------------------------------------------------------
__global__ __launch_bounds__(128) void afno_stage1(
    const float* __restrict__ x, const float* __restrict__ ws,
    float* __restrict__ s1) {
  __shared__ float bt[64 * 64];
  const int bhw = blockIdx.x;
  STAGE_TILE_2D(bt, x + (size_t)bhw * 4096, 4096, 1, 4096, 128);
  const int lane = threadIdx.x & 31, wv = threadIdx.x >> 5;
  const int half = lane >> 4, lq = lane & 15;
  const int n0 = wv * 16;
  v8f acc = gemm_acc(ws + OFF_FL + lq * 64, bt, n0 + lq, 64, half);
  float* o = s1 + (size_t)bhw * 1024;
#pragma unroll
  for (int r = 0; r < 8; ++r)
    o[(r + 8 * half) * 64 + n0 + lq] = acc[r];
}

// ---------------------------------------------------------------------------
// Stage 2: complex DFT along W (keep 32 modes). blocks = B*H*T, 256 thr.
// TDM tile: 64 lines x 128 f32, line stride 1024 f32 -> LDS [k=128][64].
// ---------------------------------------------------------------------------
__global__ __launch_bounds__(256) void afno_stage2(
    const float* __restrict__ ws, const float* __restrict__ s1,
    float* __restrict__ s2) {
  __shared__ float bt[128 * 64];
  const int id = blockIdx.x;                  // (b*64+h)*8 + t
  const int t = id & 7;
  const size_t bh = (size_t)(id >> 3);
  STAGE_TILE_2D(bt, s1 + bh * 65536 + 2 * t * 64, 128, 64, 1024, 256);
  const int lane = threadIdx.x & 31, wv = threadIdx.x >> 5;
  const int half = lane >> 4, lq = lane & 15;
  float* Obase = s2 + (size_t)id * 4096;
  for (int ti = wv; ti < 16; ti += 8) {
    const int tr = ti >> 2, tc = ti & 3;
    v8f acc = gemm_acc(ws + OFF_AF + (tr * 16 + lq) * 128, bt, tc * 16 + lq, 128, half);
#pragma unroll
    for (int r = 0; r < 8; ++r)
      Obase[(tr * 16 + r + 8 * half) * 64 + tc * 16 + lq] = acc[r];
  }
}

// ---------------------------------------------------------------------------
// Stage 3: complex DFT along H. blocks = B*M*T. line stride 8*4096 f32.
// ---------------------------------------------------------------------------
__global__ __launch_bounds__(256) void afno_stage3(
    const float* __restrict__ ws, const float* __restrict__ s2,
    float* __restrict__ s3) {
  __shared__ float bt[128 * 64];
  const int id = blockIdx.x;                  // ((b*32)+m)*8 + t
  const int t = id & 7, bm = id >> 3;
  const int m = bm & 31, b = bm >> 5;
  STAGE_TILE_2D(bt, s2 + ((size_t)b * 512 + t) * 4096 + 2 * m * 64, 128, 64, 32768, 256);
  const int lane = threadIdx.x & 31, wv = threadIdx.x >> 5;
  const int half = lane >> 4, lq = lane & 15;
  float* Obase = s3 + ((size_t)(b * 8 + t) * 32 + m) * 4096;
  for (int ti = wv; ti < 16; ti += 8) {
    const int tr = ti >> 2, tc = ti & 3;
    v8f acc = gemm_acc(ws + OFF_AF + (tr * 16 + lq) * 128, bt, tc * 16 + lq, 128, half);
#pragma unroll
    for (int r = 0; r < 8; ++r)
      Obase[(tr * 16 + r + 8 * half) * 64 + tc * 16 + lq] = acc[r];
  }
}

// ---------------------------------------------------------------------------
// Block-diagonal complex 2-layer MLP per kept mode. blocks = B*T*M*32, 64 thr.
// ---------------------------------------------------------------------------
__global__ __launch_bounds__(64) void afno_mlp(
    const float* __restrict__ ws, float* __restrict__ s3) {
  __shared__ float xr[64], xi[64], m1r[64], m1i[64];
  const int pos = blockIdx.x;
  const int k = pos & 31, rest = pos >> 5;    // rest = (b*8+t)*32+m
  float* rowr = s3 + (size_t)rest * 4096 + (2 * k) * 64;
  float* rowi = rowr + 64;
  const int c = threadIdx.x;
  xr[c] = rowr[c];
  xi[c] = rowi[c];
  __syncthreads();
  const int n = c >> 3, o = c & 7;
  const float* W1R = ws + OFF_W1R; const float* W1I = ws + OFF_W1I;
  const float* W2R = ws + OFF_W2R; const float* W2I = ws + OFF_W2I;
  float sr = ws[OFF_B1R + c] - ws[OFF_B1I + c];
  float si = ws[OFF_B1R + c] + ws[OFF_B1I + c];
#pragma unroll
  for (int i = 0; i < 8; ++i) {
    const float wr = W1R[(n * 8 + i) * 8 + o], wi = W1I[(n * 8 + i) * 8 + o];
    const float vr = xr[n * 8 + i], vi = xi[n * 8 + i];
    sr += vr * wr - vi * wi;
    si += vi * wr + vr * wi;
  }
  m1r[c] = 0.5f * sr * (1.0f + erff(sr * 0.70710678118f));
  m1i[c] = 0.5f * si * (1.0f + erff(si * 0.70710678118f));
  __syncthreads();
  float tr2 = ws[OFF_B2R + c] - ws[OFF_B2I + c];
  float ti2 = ws[OFF_B2R + c] + ws[OFF_B2I + c];
#pragma unroll
  for (int i = 0; i < 8; ++i) {
    const float wr = W2R[(n * 8 + i) * 8 + o], wi = W2I[(n * 8 + i) * 8 + o];
    const float vr = m1r[n * 8 + i], vi = m1i[n * 8 + i];
    tr2 += vr * wr - vi * wi;
    ti2 += vi * wr + vr * wi;
  }
  rowr[c] = tr2;
  rowi[c] = ti2;
}

// ---------------------------------------------------------------------------
// Stage 4: inverse DFT along H. blocks = B*M*T. Contiguous 16KB B tile.
// Scatter into S4(=S2 space) [b][h][t][2m+p][c].
// ---------------------------------------------------------------------------
__global__ __launch_bounds__(256) void afno_stage4(
    const float* __restrict__ ws, const float* __restrict__ s3,
    float* __restrict__ s4) {
  __shared__ float bt[64 * 64];
  const int id = blockIdx.x;
  const int t = id & 7, bm = id >> 3;
  const int m = bm & 31, b = bm >> 5;
  STAGE_TILE_2D(bt, s3 + ((size_t)(b * 8 + t) * 32 + m) * 4096, 4096, 1, 4096, 256);
  const int lane = threadIdx.x & 31, wv = threadIdx.x >> 5;
  const int half = lane >> 4, lq = lane & 15;
  for (int ti = wv; ti < 32; ti += 8) {
    const int tr = ti >> 2, tc = ti & 3;
    v8f acc = gemm_acc(ws + OFF_AI + (tr * 16 + lq) * 64, bt, tc * 16 + lq, 64, half);
#pragma unroll
    for (int r = 0; r < 8; ++r) {
      const int h2 = tr * 16 + r + 8 * half;
      const int h = h2 >> 1, p = h2 & 1;
      s4[((size_t)(b * 64 + h) * 8 + t) * 4096 + (2 * m + p) * 64 + tc * 16 + lq] = acc[r];
    }
  }
}

// ---------------------------------------------------------------------------
// Stage 5: inverse DFT along W. blocks = B*H*T. S4(=S2) -> S5(=S1 space).
// ---------------------------------------------------------------------------
__global__ __launch_bounds__(256) void afno_stage5(
    const float* __restrict__ ws, const float* __restrict__ s4,
    float* __restrict__ s5) {
  __shared__ float bt[64 * 64];
  const int id = blockIdx.x;                  // (b*64+h)*8 + t
  const int t = id & 7;
  const size_t bh = (size_t)(id >> 3);
  STAGE_TILE_2D(bt, s4 + (size_t)id * 4096, 4096, 1, 4096, 256);
  const int lane = threadIdx.x & 31, wv = threadIdx.x >> 5;
  const int half = lane >> 4, lq = lane & 15;
  for (int ti = wv; ti < 32; ti += 8) {
    const int tr = ti >> 2, tc = ti & 3;
    v8f acc = gemm_acc(ws + OFF_AI + (tr * 16 + lq) * 64, bt, tc * 16 + lq, 64, half);
#pragma unroll
    for (int r = 0; r < 8; ++r) {
      const int w2 = tr * 16 + r + 8 * half;
      const int w = w2 >> 1, p = w2 & 1;
      s5[(bh * 64 + w) * 1024 + (2 * t + p) * 64 + tc * 16 + lq] = acc[r];
    }
  }
}

// ---------------------------------------------------------------------------
// Stage 6: hermitian inverse rDFT along L + residual. blocks = B*H*W.
// [64 x 16] @ [16 x 64]; out = D + x.
// ---------------------------------------------------------------------------
__global__ __launch_bounds__(256) void afno_stage6(
    const float* __restrict__ ws, const float* __restrict__ s5,
    const float* __restrict__ x, float* __restrict__ out) {
  __shared__ float bt[16 * 64];
  const int bhw = blockIdx.x;
  STAGE_TILE_2D(bt, s5 + (size_t)bhw * 1024, 1024, 1, 1024, 256);
  const int lane = threadIdx.x & 31, wv = threadIdx.x >> 5;
  const int half = lane >> 4, lq = lane & 15;
  for (int ti = wv; ti < 16; ti += 8) {
    const int tr = ti >> 2, tc = ti & 3;
    v8f acc = gemm_acc(ws + OFF_GL + (tr * 16 + lq) * 16, bt, tc * 16 + lq, 16, half);
#pragma unroll
    for (int r = 0; r < 8; ++r) {
      const int l = tr * 16 + r + 8 * half;
      const size_t addr = (size_t)bhw * 4096 + (size_t)l * 64 + tc * 16 + lq;
      out[addr] = acc[r] + x[addr];
    }
  }
}

// ---------------------------------------------------------------------------
extern "C" void kernel_launch(void* const* d_in, const int* in_sizes, int n_in,
                              void* d_out, int out_size, void* d_ws, size_t ws_size,
                              hipStream_t stream) {
  (void)in_sizes; (void)n_in; (void)out_size; (void)ws_size;
  const float* x     = (const float*)d_in[0];
  const float* w1    = (const float*)d_in[1];
  const float* b1    = (const float*)d_in[2];
  const float* w2    = (const float*)d_in[3];
  const float* b2    = (const float*)d_in[4];
  const float* la1rA = (const float*)d_in[5];
  const float* la1rB = (const float*)d_in[6];
  const float* la1iA = (const float*)d_in[7];
  const float* la1iB = (const float*)d_in[8];
  const float* la2rA = (const float*)d_in[9];
  const float* la2rB = (const float*)d_in[10];
  const float* la2iA = (const float*)d_in[11];
  const float* la2iB = (const float*)d_in[12];
  float* out = (float*)d_out;

  float* ws = (float*)d_ws;
  float* s1 = ws + OFF_S1;   // also S5
  float* s2 = ws + OFF_S2;   // also S4
  float* s3 = ws + OFF_S3;

  afno_init_tables<<<1, 256, 0, stream>>>(w1, b1, w2, b2,
      la1rA, la1rB, la1iA, la1iB, la2rA, la2rB, la2iA, la2iB, ws);
  afno_stage1<<<Bn * Hn * Wn, 128, 0, stream>>>(x, ws, s1);
  afno_stage2<<<Bn * Hn * Tt, 256, 0, stream>>>(ws, s1, s2);
  afno_stage3<<<Bn * Mm * Tt, 256, 0, stream>>>(ws, s2, s3);
  afno_mlp<<<Bn * Tt * Mm * 32, 64, 0, stream>>>(ws, s3);
  afno_stage4<<<Bn * Mm * Tt, 256, 0, stream>>>(ws, s3, s2);   // s4 aliases s2
  afno_stage5<<<Bn * Hn * Tt, 256, 0, stream>>>(ws, s2, s1);   // s5 aliases s1
  afno_stage6<<<Bn * Hn * Wn, 256, 0, stream>>>(ws, s1, x, out);
}